// SparseGrid_54125177864604
// MI455X (gfx1250) — compile-verified
//
#include <hip/hip_runtime.h>

// Trilinear sparse-grid sampling (Plenoxels sample_grid, want_colors=True).
// Memory-bound gather kernel tuned for MI455X:
//  - gather tables (links 8MB + density 6.4MB + sh 172.8MB ~ 187MB) ~ fit the
//    192MB L2 -> keep those accesses temporal (default)
//  - points (read-once) and outputs (write-once, 224MB) use non-temporal
//    hints so they don't evict the L2-resident tables
//  - points and out_sh staged through LDS for fully coalesced global traffic
//  - global_prefetch_b8 on the 8 gathered SH rows right after links resolve

#define BLOCK 256
#define N_SH 27

__global__ __launch_bounds__(BLOCK)
void sample_grid_kernel(const float* __restrict__ density,
                        const float* __restrict__ sh,
                        const int*   __restrict__ links,
                        const float* __restrict__ points,
                        float* __restrict__ out_d,
                        float* __restrict__ out_sh,
                        int n)
{
    __shared__ float s_pts[BLOCK * 3];
    __shared__ float s_sh [BLOCK * N_SH];

    const int tid        = threadIdx.x;
    const int block_base = blockIdx.x * BLOCK;
    const int i          = block_base + tid;

    // ---- stage this block's points through LDS (coalesced, non-temporal) ----
    {
        const int pbase = block_base * 3;
        const int plim  = n * 3;
#pragma unroll
        for (int j = 0; j < 3; ++j) {
            const int g = pbase + j * BLOCK + tid;
            float v = 0.0f;
            if (g < plim) v = __builtin_nontemporal_load(points + g);
            s_pts[j * BLOCK + tid] = v;
        }
    }
    __syncthreads();

    float    wt[8];
    unsigned rowoff[8];           // SH row offsets (floats); byte offset < 2^31
#pragma unroll
    for (int k = 0; k < 8; ++k) { wt[k] = 0.0f; rowoff[k] = 0u; }
    float dsum = 0.0f;

    if (i < n) {
        float px = fminf(fmaxf(s_pts[tid * 3 + 0], 0.0f), 127.0f);
        float py = fminf(fmaxf(s_pts[tid * 3 + 1], 0.0f), 127.0f);
        float pz = fminf(fmaxf(s_pts[tid * 3 + 2], 0.0f), 127.0f);

        int lx = (int)floorf(px); lx = lx < 0 ? 0 : (lx > 126 ? 126 : lx);
        int ly = (int)floorf(py); ly = ly < 0 ? 0 : (ly > 126 ? 126 : ly);
        int lz = (int)floorf(pz); lz = lz < 0 ? 0 : (lz > 126 ? 126 : lz);

        const float wx = px - (float)lx, wy = py - (float)ly, wz = pz - (float)lz;
        const float ox = 1.0f - wx,      oy = 1.0f - wy,      oz = 1.0f - wz;

        // links is [128,128,128] row-major, z fastest
        const int base = (lx << 14) + (ly << 7) + lz;
        int lnk[8];
        lnk[0] = links[base];
        lnk[1] = links[base + 1];
        lnk[2] = links[base + 128];
        lnk[3] = links[base + 129];
        lnk[4] = links[base + 16384];
        lnk[5] = links[base + 16385];
        lnk[6] = links[base + 16512];
        lnk[7] = links[base + 16513];

        const float wxs[2] = {ox, wx}, wys[2] = {oy, wy}, wzs[2] = {oz, wz};
#pragma unroll
        for (int k = 0; k < 8; ++k) {
            const int dx = (k >> 2) & 1, dy = (k >> 1) & 1, dz = k & 1;
            const int   v = lnk[k];
            const float w = wxs[dx] * wys[dy] * wzs[dz];
            wt[k]     = (v >= 0) ? w : 0.0f;
            const unsigned s = (unsigned)((v >= 0) ? v : 0);
            rowoff[k] = s * 27u;
            // start pulling the SH row toward L2/WGP$ now (global_prefetch_b8)
            __builtin_prefetch(sh + rowoff[k], 0, 0);
            dsum += wt[k] * density[s];
        }
    }

    // density output is naturally coalesced; stream it (write-once)
    if (i < n) __builtin_nontemporal_store(dsum, out_d + i);

    // ---- 27 SH channels: gather-FMA into LDS (inactive lanes write zeros:
    //      wt==0, rowoff==0 keeps the dummy loads in-bounds) ----
#pragma unroll 9
    for (int c = 0; c < N_SH; ++c) {
        float acc;
        acc  = wt[0] * sh[rowoff[0] + c];
        acc += wt[1] * sh[rowoff[1] + c];
        acc += wt[2] * sh[rowoff[2] + c];
        acc += wt[3] * sh[rowoff[3] + c];
        acc += wt[4] * sh[rowoff[4] + c];
        acc += wt[5] * sh[rowoff[5] + c];
        acc += wt[6] * sh[rowoff[6] + c];
        acc += wt[7] * sh[rowoff[7] + c];
        s_sh[tid * N_SH + c] = acc;     // stride 27 vs 64 banks: conflict-free
    }
    __syncthreads();

    // ---- coalesced, non-temporal writeback of the block's SH tile ----
    int npts = n - block_base;
    if (npts > BLOCK) npts = BLOCK;
    const int total = npts * N_SH;                      // contiguous region
    float* dst = out_sh + (size_t)block_base * N_SH;
    for (int j = tid; j < total; j += BLOCK)
        __builtin_nontemporal_store(s_sh[j], dst + j);
}

extern "C" void kernel_launch(void* const* d_in, const int* in_sizes, int n_in,
                              void* d_out, int out_size, void* d_ws, size_t ws_size,
                              hipStream_t stream) {
    const float* density = (const float*)d_in[0];   // [CAP, 1]
    const float* sh      = (const float*)d_in[1];   // [CAP, 27]
    const int*   links   = (const int*)  d_in[2];   // [128,128,128]
    const float* points  = (const float*)d_in[3];   // [N, 3]
    const int n = in_sizes[3] / 3;

    float* out_d  = (float*)d_out;                  // [N, 1]
    float* out_sh = out_d + n;                      // [N, 27]

    const int grid = (n + BLOCK - 1) / BLOCK;
    sample_grid_kernel<<<grid, BLOCK, 0, stream>>>(density, sh, links, points,
                                                   out_d, out_sh, n);
}